// SparseGraphTransformerAttention_80272938763030
// MI455X (gfx1250) — compile-verified
//
#include <hip/hip_runtime.h>
#include <math.h>

// ---------------------------------------------------------------------------
// SparseGraphTransformerAttention for MI455X (gfx1250, wave32, WMMA)
//
// Memory-bound problem (~120 MB traffic vs 1.23 GFLOP of GEMM), so:
//  - fp32 end-to-end (no precision sacrifice; bandwidth is the ceiling)
//  - projections via V_WMMA_F32_16X16X4_F32 (native fp32 WMMA)
//  - q/k tables (25.6 MB) stay resident in the 192 MB L2 for the edge gather
//  - segment max via orderable-uint atomicMax keys; segment sum via atomicAdd
// ---------------------------------------------------------------------------

typedef __attribute__((ext_vector_type(2))) float v2f;
typedef __attribute__((ext_vector_type(8))) float v8f;

#define F      64   // IN_F == D_ATT
#define HEADS  8
#define DH     8    // D_HEAD
#define INV_SQRT_DH 0.35355339059327373f

// Monotone map float -> uint so unsigned atomicMax == float max (incl. negatives).
__device__ __forceinline__ unsigned fkey(float f) {
  unsigned u = __float_as_uint(f);
  return (u & 0x80000000u) ? ~u : (u | 0x80000000u);
}
__device__ __forceinline__ float funkey(unsigned u) {
  return __uint_as_float((u & 0x80000000u) ? (u & 0x7FFFFFFFu) : ~u);
}

// -------------------------------------------------------------------- init --
__global__ void init_seg(unsigned* __restrict__ smax, float* __restrict__ ssum, int n) {
  int i = blockIdx.x * blockDim.x + threadIdx.x;
  if (i < n) {
    smax[i] = 0x007FFFFFu;  // fkey(-inf): empty-segment sentinel
    ssum[i] = 0.0f;
  }
}

// ------------------------------------------------- q/k/v projection (WMMA) --
// Grid: N/16 blocks of 128 threads (4 waves). Block owns rows [16*bid, +16).
// Wave w owns output columns [16*w, +16). K loop: 16 steps of K=4, 3 WMMAs
// per step (one shared A fragment, three B fragments for Wq/Wk/Wv).
//
// ISA 7.12.2 fragment layouts (wave32, 32-bit):
//   A 16x4 : lanes 0-15 rows M=lane, VGPR0/1 = K=0/1; lanes 16-31 K=2/3.
//   B 4x16 : VGPR0 = rows K=0 (lanes 0-15) / K=2 (lanes 16-31); VGPR1 = K=1/K=3.
//   C 16x16: VGPR r = row M=r (lanes 0-15) / M=r+8 (lanes 16-31), N=lane%16.
__global__ __launch_bounds__(128)
void proj_qkv(const float* __restrict__ x,
              const float* __restrict__ Bq, const float* __restrict__ biasq,
              const float* __restrict__ Bk, const float* __restrict__ biask,
              const float* __restrict__ Bv, const float* __restrict__ biasv,
              float* __restrict__ qo, float* __restrict__ ko, float* __restrict__ vo) {
  const int lane    = threadIdx.x & 31;
  const int wave    = threadIdx.x >> 5;        // column tile 0..3
  const int rowBase = blockIdx.x * 16;
  const int colBase = wave * 16;
  const int half    = lane >> 4;               // 0 = lanes 0-15, 1 = lanes 16-31
  const int lr      = lane & 15;

  const int arow = rowBase + lr;               // A-fragment row for this lane
  const int col  = colBase + lr;               // B/C column for this lane

  v8f cq = {0.f,0.f,0.f,0.f,0.f,0.f,0.f,0.f};
  v8f ck = cq;
  v8f cv = cq;

  #pragma unroll
  for (int kk = 0; kk < F; kk += 4) {
    const int ka = kk + half * 2;              // this lane's K pair
    // A fragment: contiguous float2 from x (row stride 64, 8B aligned)
    const float2 av = *(const float2*)(x + arow * F + ka);
    v2f a; a.x = av.x; a.y = av.y;

    v2f bq, bk, bv;
    bq.x = Bq[(ka + 0) * F + col]; bq.y = Bq[(ka + 1) * F + col];
    bk.x = Bk[(ka + 0) * F + col]; bk.y = Bk[(ka + 1) * F + col];
    bv.x = Bv[(ka + 0) * F + col]; bv.y = Bv[(ka + 1) * F + col];

    cq = __builtin_amdgcn_wmma_f32_16x16x4_f32(false, a, false, bq, (short)0, cq, false, false);
    ck = __builtin_amdgcn_wmma_f32_16x16x4_f32(false, a, false, bk, (short)0, ck, false, false);
    cv = __builtin_amdgcn_wmma_f32_16x16x4_f32(false, a, false, bv, (short)0, cv, false, false);
  }

  const float addq = biasq[col];
  const float addk = biask[col];
  const float addv = biasv[col];
  const int   orow = rowBase + half * 8;
  #pragma unroll
  for (int r = 0; r < 8; ++r) {
    const int o = (orow + r) * F + col;
    qo[o] = cq[r] + addq;
    ko[o] = ck[r] + addk;
    vo[o] = cv[r] + addv;
  }
}

// ------------------------------------- edge products + segment-max scatter --
// prods[e,d] = (1/sqrt(8)) * sum_h q[src][h*8+d] * k[dst][h*8+d]
__global__ __launch_bounds__(256)
void edge_prods(const float* __restrict__ q, const float* __restrict__ k,
                const int* __restrict__ src, const int* __restrict__ dst,
                float* __restrict__ prods, unsigned* __restrict__ smax, int E) {
  const int e = blockIdx.x * blockDim.x + threadIdx.x;
  if (e >= E) return;
  const int s = src[e];
  const int d = dst[e];
  const float4* qr = (const float4*)(q + (size_t)s * F);
  const float4* kr = (const float4*)(k + (size_t)d * F);

  float acc[DH] = {0.f,0.f,0.f,0.f,0.f,0.f,0.f,0.f};
  #pragma unroll
  for (int h = 0; h < HEADS; ++h) {
    const float4 qa = qr[2 * h + 0], qb = qr[2 * h + 1];
    const float4 ka = kr[2 * h + 0], kb = kr[2 * h + 1];
    acc[0] = fmaf(qa.x, ka.x, acc[0]);
    acc[1] = fmaf(qa.y, ka.y, acc[1]);
    acc[2] = fmaf(qa.z, ka.z, acc[2]);
    acc[3] = fmaf(qa.w, ka.w, acc[3]);
    acc[4] = fmaf(qb.x, kb.x, acc[4]);
    acc[5] = fmaf(qb.y, kb.y, acc[5]);
    acc[6] = fmaf(qb.z, kb.z, acc[6]);
    acc[7] = fmaf(qb.w, kb.w, acc[7]);
  }

  float4 p0 = make_float4(acc[0] * INV_SQRT_DH, acc[1] * INV_SQRT_DH,
                          acc[2] * INV_SQRT_DH, acc[3] * INV_SQRT_DH);
  float4 p1 = make_float4(acc[4] * INV_SQRT_DH, acc[5] * INV_SQRT_DH,
                          acc[6] * INV_SQRT_DH, acc[7] * INV_SQRT_DH);
  float4* pr = (float4*)(prods + (size_t)e * DH);
  pr[0] = p0;
  pr[1] = p1;

  unsigned* sm = smax + (size_t)d * DH;
  atomicMax(&sm[0], fkey(p0.x)); atomicMax(&sm[1], fkey(p0.y));
  atomicMax(&sm[2], fkey(p0.z)); atomicMax(&sm[3], fkey(p0.w));
  atomicMax(&sm[4], fkey(p1.x)); atomicMax(&sm[5], fkey(p1.y));
  atomicMax(&sm[6], fkey(p1.z)); atomicMax(&sm[7], fkey(p1.w));
}

// ---------------------------------------------- exp + segment-sum scatter --
__global__ __launch_bounds__(256)
void edge_expsum(const float* __restrict__ prods, const int* __restrict__ dst,
                 const unsigned* __restrict__ smax, float* __restrict__ ssum, int E) {
  const int e = blockIdx.x * blockDim.x + threadIdx.x;
  if (e >= E) return;
  const int d = dst[e];
  const float4* pr = (const float4*)(prods + (size_t)e * DH);
  const float4 p0 = pr[0], p1 = pr[1];
  const unsigned* sm = smax + (size_t)d * DH;
  float*          ss = (float*)(ssum + (size_t)d * DH);

  atomicAdd(&ss[0], expf(p0.x - funkey(sm[0])));
  atomicAdd(&ss[1], expf(p0.y - funkey(sm[1])));
  atomicAdd(&ss[2], expf(p0.z - funkey(sm[2])));
  atomicAdd(&ss[3], expf(p0.w - funkey(sm[3])));
  atomicAdd(&ss[4], expf(p1.x - funkey(sm[4])));
  atomicAdd(&ss[5], expf(p1.y - funkey(sm[5])));
  atomicAdd(&ss[6], expf(p1.z - funkey(sm[6])));
  atomicAdd(&ss[7], expf(p1.w - funkey(sm[7])));
}

// ------------------------------------------------------ final attention ----
__global__ __launch_bounds__(256)
void edge_att(const float* __restrict__ prods, const int* __restrict__ dst,
              const unsigned* __restrict__ smax, const float* __restrict__ ssum,
              float* __restrict__ att, int E) {
  const int e = blockIdx.x * blockDim.x + threadIdx.x;
  if (e >= E) return;
  const int d = dst[e];
  const float4* pr = (const float4*)(prods + (size_t)e * DH);
  const float4 p0 = pr[0], p1 = pr[1];
  const unsigned* sm = smax + (size_t)d * DH;
  const float*    ss = ssum + (size_t)d * DH;

  float4 a0, a1;
  a0.x = expf(p0.x - funkey(sm[0])) / (ss[0] + 1e-16f);
  a0.y = expf(p0.y - funkey(sm[1])) / (ss[1] + 1e-16f);
  a0.z = expf(p0.z - funkey(sm[2])) / (ss[2] + 1e-16f);
  a0.w = expf(p0.w - funkey(sm[3])) / (ss[3] + 1e-16f);
  a1.x = expf(p1.x - funkey(sm[4])) / (ss[4] + 1e-16f);
  a1.y = expf(p1.y - funkey(sm[5])) / (ss[5] + 1e-16f);
  a1.z = expf(p1.z - funkey(sm[6])) / (ss[6] + 1e-16f);
  a1.w = expf(p1.w - funkey(sm[7])) / (ss[7] + 1e-16f);

  float4* ao = (float4*)(att + (size_t)e * DH);
  ao[0] = a0;
  ao[1] = a1;
}

// ---------------------------------------------------------------------------
extern "C" void kernel_launch(void* const* d_in, const int* in_sizes, int n_in,
                              void* d_out, int out_size, void* d_ws, size_t ws_size,
                              hipStream_t stream) {
  const float* x     = (const float*)d_in[0];
  const float* Wq    = (const float*)d_in[1];
  const float* biasq = (const float*)d_in[2];
  const float* Wk    = (const float*)d_in[3];
  const float* biask = (const float*)d_in[4];
  const float* Wv    = (const float*)d_in[5];
  const float* biasv = (const float*)d_in[6];
  const int*   edge  = (const int*)d_in[7];

  const int N = in_sizes[0] / F;   // 50000 (divisible by 16)
  const int E = in_sizes[7] / 2;   // 800000
  const int* src = edge;           // edge[0]
  const int* dst = edge + E;       // edge[1]

  // Output tuple layout: attention [E,8] | v [N,64] | prods [E,8]
  float* att = (float*)d_out;
  float* vo  = att + (size_t)E * DH;
  float* pr  = vo + (size_t)N * F;

  // Workspace: q [N,64] | k [N,64] | segmax keys [N,8] | segsum [N,8]  (~28.8 MB)
  float*    qws  = (float*)d_ws;
  float*    kws  = qws + (size_t)N * F;
  unsigned* smax = (unsigned*)(kws + (size_t)N * F);
  float*    ssum = (float*)(smax + (size_t)N * DH);

  const int nseg = N * DH;
  init_seg<<<(nseg + 255) / 256, 256, 0, stream>>>(smax, ssum, nseg);

  proj_qkv<<<N / 16, 128, 0, stream>>>(x, Wq, biasq, Wk, biask, Wv, biasv,
                                       qws, kws, vo);

  const int eblk = (E + 255) / 256;
  edge_prods <<<eblk, 256, 0, stream>>>(qws, kws, src, dst, pr, smax, E);
  edge_expsum<<<eblk, 256, 0, stream>>>(pr, dst, smax, ssum, E);
  edge_att   <<<eblk, 256, 0, stream>>>(pr, dst, smax, ssum, att, E);
}